// ElasticTransform_55903294324925
// MI455X (gfx1250) — compile-verified
//
#include <hip/hip_runtime.h>
#include <math.h>

#define KTAPS 97
#define PAD   48
#define NI    32
#define HH    512
#define WW    512
#define HWSZ  (HH * WW)

typedef __attribute__((ext_vector_type(2))) float v2f;
typedef __attribute__((ext_vector_type(8))) float v8f;

// ---------------------------------------------------------------------------
// Kernel 0: build normalized Gaussian taps g[2][97] from log_sigma.
// ---------------------------------------------------------------------------
__global__ __launch_bounds__(256) void prep_gauss(const float* __restrict__ log_sigma,
                                                  float* __restrict__ g_ws) {
    __shared__ float raw[2][128];
    __shared__ float inv[2];
    const int t  = threadIdx.x;      // 0..255
    const int ch = t >> 7;
    const int k  = t & 127;
    float val = 0.0f;
    if (k < KTAPS) {
        float s  = expf(log_sigma[ch]);
        float xk = (float)(k - PAD);
        val = expf(-(xk * xk) / (2.0f * s * s));
    }
    raw[ch][k] = val;
    __syncthreads();
    if (k == 0) {
        float s = 0.0f;
        for (int i = 0; i < KTAPS; ++i) s += raw[ch][i];
        inv[ch] = 1.0f / s;
    }
    __syncthreads();
    if (k < KTAPS) g_ws[ch * KTAPS + k] = val * inv[ch];
}

// ---------------------------------------------------------------------------
// Kernel 1: vertical 97-tap blur of d = 2*d_rand - 1, via FP32 WMMA.
// One wave -> one 16x16 output tile. Out[m][n] = sum_r g[r-m] * tile[r][n],
// r in [0,112). A = Toeplitz chunk (16x4 of g), B = input chunk (4x16).
// ---------------------------------------------------------------------------
__global__ __launch_bounds__(256) void blur_v(const float* __restrict__ drand,
                                              float* __restrict__ tmp,
                                              const float* __restrict__ g_ws) {
    __shared__ float gpad[128];            // g with 16-entry zero skirt: gpad[t+16]=g[t]
    __shared__ float tile[8][112 * 16];    // per-wave input strip (rows h0-48..h0+63)

    const int lane = threadIdx.x & 31;
    const int wv   = threadIdx.x >> 5;
    const int wid  = blockIdx.x * 8 + wv;
    const int tw   = wid & 31;             // tile col
    const int th   = (wid >> 5) & 31;      // tile row
    const int img  = wid >> 10;            // n*2 + ch  (uniform per block: 1024 tiles/plane)
    const int ch   = img & 1;
    const int h0   = th * 16;
    const int w0   = tw * 16;

    if (threadIdx.x < 128) {
        int tap = (int)threadIdx.x - 16;
        float v = 0.0f;
        if (tap >= 0 && tap < KTAPS) v = g_ws[ch * KTAPS + tap];
        gpad[threadIdx.x] = v;
    }
    const float* src = drand + (size_t)img * HWSZ;
    for (int i = lane; i < 112 * 16; i += 32) {
        int r  = i >> 4;
        int c  = i & 15;
        int hi = h0 - PAD + r;
        float v = 0.0f;
        if (hi >= 0 && hi < HH) v = 2.0f * src[hi * WW + w0 + c] - 1.0f;
        tile[wv][i] = v;
    }
    __syncthreads();

    const int m  = lane & 15;   // A: row M. B/D: col N.
    const int hs = lane >> 4;   // half select: VGPR0 holds K={0,2}, VGPR1 K={1,3}
    const float* tl = tile[wv];
    v8f acc = {};
#pragma unroll 4
    for (int ki = 0; ki < 28; ++ki) {
        const int base = 4 * ki + 2 * hs;
        v2f a, b;
        a.x = gpad[base     - m + 16];     // Toeplitz: g[r - m], zero-padded, in-bounds
        a.y = gpad[base + 1 - m + 16];
        b.x = tl[ base      * 16 + m];     // input rows base..base+1, col m
        b.y = tl[(base + 1) * 16 + m];
        acc = __builtin_amdgcn_wmma_f32_16x16x4_f32(
            false, a, false, b, (short)0, acc, false, false);
    }

    float* dst = tmp + (size_t)img * HWSZ + (size_t)h0 * WW + w0;
#pragma unroll
    for (int v = 0; v < 8; ++v)
        dst[(v + 8 * hs) * WW + m] = acc[v];   // C/D: VGPR v -> row M=v(+8), lane -> col N
}

// ---------------------------------------------------------------------------
// Kernel 2: horizontal 97-tap blur + alpha scale, via FP32 WMMA.
// Out[m][n] = sum_c in[h0+m][w0-48+c] * g[c-n] : A = image (16x4), B = Toeplitz.
// LDS pitch 113 => column reads hit 16 distinct banks (49*r mod 64).
// ---------------------------------------------------------------------------
__global__ __launch_bounds__(256) void blur_h(const float* __restrict__ tmp,
                                              float* __restrict__ disp,
                                              const float* __restrict__ g_ws,
                                              const float* __restrict__ log_alpha) {
    __shared__ float gpad[128];
    __shared__ float tile[8][16 * 113];

    const int lane = threadIdx.x & 31;
    const int wv   = threadIdx.x >> 5;
    const int wid  = blockIdx.x * 8 + wv;
    const int tw   = wid & 31;
    const int th   = (wid >> 5) & 31;
    const int img  = wid >> 10;
    const int ch   = img & 1;
    const int h0   = th * 16;
    const int w0   = tw * 16;

    if (threadIdx.x < 128) {
        int tap = (int)threadIdx.x - 16;
        float v = 0.0f;
        if (tap >= 0 && tap < KTAPS) v = g_ws[ch * KTAPS + tap];
        gpad[threadIdx.x] = v;
    }
    const float alpha = expf(log_alpha[ch]);
    const float* src = tmp + (size_t)img * HWSZ;
    for (int i = lane; i < 16 * 112; i += 32) {
        int r  = i / 112;
        int c  = i - r * 112;
        int wi = w0 - PAD + c;
        float v = 0.0f;
        if (wi >= 0 && wi < WW) v = src[(h0 + r) * WW + wi];
        tile[wv][r * 113 + c] = v;
    }
    __syncthreads();

    const int m  = lane & 15;
    const int hs = lane >> 4;
    const float* tl = tile[wv];
    v8f acc = {};
#pragma unroll 4
    for (int ki = 0; ki < 28; ++ki) {
        const int base = 4 * ki + 2 * hs;
        v2f a, b;
        a.x = tl[m * 113 + base];          // image row m, cols base..base+1
        a.y = tl[m * 113 + base + 1];
        b.x = gpad[base     - m + 16];     // Toeplitz g[c - n] (lane = col N here)
        b.y = gpad[base + 1 - m + 16];
        acc = __builtin_amdgcn_wmma_f32_16x16x4_f32(
            false, a, false, b, (short)0, acc, false, false);
    }

    float* dst = disp + (size_t)img * HWSZ + (size_t)h0 * WW + w0;
#pragma unroll
    for (int v = 0; v < 8; ++v)
        dst[(v + 8 * hs) * WW + m] = acc[v] * alpha;
}

// ---------------------------------------------------------------------------
// Kernel 3: bilinear warp. One thread per (n,h,w); loops over 3 channels.
// ---------------------------------------------------------------------------
__global__ __launch_bounds__(256) void warp_bilinear(const float* __restrict__ x,
                                                     const float* __restrict__ disp,
                                                     float* __restrict__ out) {
    const unsigned tid = blockIdx.x * 256u + threadIdx.x;   // < 32*2^18
    const int n  = tid >> 18;
    const int hw = tid & (HWSZ - 1);
    const int h  = hw >> 9;
    const int w  = hw & 511;

    const float d0 = disp[((size_t)(n * 2 + 0)) * HWSZ + hw];  // perturbs x coord
    const float d1 = disp[((size_t)(n * 2 + 1)) * HWSZ + hw];  // perturbs y coord

    float gx = -1.0f + 2.0f * (float)w / (float)(WW - 1) + d0;
    float gy = -1.0f + 2.0f * (float)h / (float)(HH - 1) + d1;
    gx = fminf(fmaxf(gx, -1.0f), 1.0f);
    gy = fminf(fmaxf(gy, -1.0f), 1.0f);
    const float ix = (gx + 1.0f) * 0.5f * (float)(WW - 1);
    const float iy = (gy + 1.0f) * 0.5f * (float)(HH - 1);
    const float x0f = floorf(ix), y0f = floorf(iy);
    const float wx = ix - x0f,   wy = iy - y0f;
    const int x0 = min(max((int)x0f, 0), WW - 1);
    const int y0 = min(max((int)y0f, 0), HH - 1);
    const int x1 = min(x0 + 1, WW - 1);
    const int y1 = min(y0 + 1, HH - 1);
    const float w00 = (1.0f - wy) * (1.0f - wx);
    const float w01 = (1.0f - wy) * wx;
    const float w10 = wy * (1.0f - wx);
    const float w11 = wy * wx;

    const float* xb = x   + (size_t)n * 3 * HWSZ;
    float*       ob = out + (size_t)n * 3 * HWSZ;
#pragma unroll
    for (int c = 0; c < 3; ++c) {
        const float* xc = xb + (size_t)c * HWSZ;
        float v00 = xc[y0 * WW + x0];
        float v01 = xc[y0 * WW + x1];
        float v10 = xc[y1 * WW + x0];
        float v11 = xc[y1 * WW + x1];
        ob[(size_t)c * HWSZ + hw] = v00 * w00 + v01 * w01 + v10 * w10 + v11 * w11;
    }
}

// ---------------------------------------------------------------------------
extern "C" void kernel_launch(void* const* d_in, const int* in_sizes, int n_in,
                              void* d_out, int out_size, void* d_ws, size_t ws_size,
                              hipStream_t stream) {
    (void)in_sizes; (void)n_in; (void)out_size; (void)ws_size;
    const float* x         = (const float*)d_in[0];   // (32,3,512,512)
    const float* d_rand    = (const float*)d_in[1];   // (32,2,512,512)
    const float* log_sigma = (const float*)d_in[2];   // (2,)
    const float* log_alpha = (const float*)d_in[3];   // (2,)
    float* out = (float*)d_out;

    float* g_ws = (float*)d_ws;                 // 2*97 taps (padded to 256 floats)
    float* disp = (float*)d_ws + 256;           // 32*2*512*512 floats (~67 MB)
    float* tmp  = out;                          // vertical-pass result reuses d_out

    prep_gauss<<<1, 256, 0, stream>>>(log_sigma, g_ws);

    const int conv_blocks = (NI * 2 * (HH / 16) * (WW / 16)) / 8;   // 8192
    blur_v<<<conv_blocks, 256, 0, stream>>>(d_rand, tmp, g_ws);
    blur_h<<<conv_blocks, 256, 0, stream>>>(tmp, disp, g_ws, log_alpha);

    const int warp_blocks = (NI * HWSZ) / 256;                      // 32768
    warp_bilinear<<<warp_blocks, 256, 0, stream>>>(x, disp, out);
}